// AC_MSA_7103875907883
// MI455X (gfx1250) — compile-verified
//
#include <hip/hip_runtime.h>
#include <hip/hip_bf16.h>
#include <math.h>

// ---------------------------------------------------------------------------
// AC-MSA (category-sorted block attention) for MI455X / gfx1250.
// fp32 throughout (matches reference numerics) using V_WMMA_F32_16X16X4_F32.
// b=4, n=16384, c=192, nh=6, d=32, group=128 -> ng=128, pad_n=0.
// v2: operand-fragment reuse -> ~1 LDS pair per WMMA; float4 softmax.
// ---------------------------------------------------------------------------

typedef __attribute__((ext_vector_type(2))) float v2f;
typedef __attribute__((ext_vector_type(8))) float v8f;

#define NHEADS 6
#define CDIM   192
#define DHEAD  32
#define GSIZE  128
#define NBATCH 4
#define NTOK   16384      // tokens per batch (128*128)
#define QSTR   36         // LDS row stride for 128x32 tiles (16B aligned, conflict-free)
#define SSTR   132        // LDS row stride for 128x128 score tile (16B aligned, 4-bank row shift)

static __device__ __forceinline__ v8f wmma4(v2f a, v2f b, v8f c) {
  // D = A(16x4) * B(4x16) + C(16x16), fp32.
  // args: (neg_a, A, neg_b, B, c_mod, C, reuse_a, reuse_b)
  return __builtin_amdgcn_wmma_f32_16x16x4_f32(false, a, false, b, (short)0, c,
                                               false, false);
}

#define V8F_ZERO ((v8f){0.f, 0.f, 0.f, 0.f, 0.f, 0.f, 0.f, 0.f})

// ---------------------------------------------------------------------------
// Kernel 1: stable counting sort by category (matches stable jnp.argsort).
// One 256-thread workgroup per batch. order[b][pos] = original token index.
// ---------------------------------------------------------------------------
__global__ __launch_bounds__(256) void acmsa_sort_kernel(
    const int* __restrict__ indices, int* __restrict__ order) {
  const int b = blockIdx.x;
  const int* cat = indices + (size_t)b * NTOK;
  int* ord = order + (size_t)b * NTOK;

  __shared__ int counts[64];
  __shared__ int running[64];
  __shared__ int ccat[256];

  const int t = threadIdx.x;
  if (t < 64) counts[t] = 0;
  __syncthreads();
  for (int i = t; i < NTOK; i += 256) atomicAdd(&counts[cat[i] & 63], 1);
  __syncthreads();
  if (t == 0) {
    int s = 0;
    for (int c = 0; c < 64; ++c) { running[c] = s; s += counts[c]; }
  }
  __syncthreads();

  const int nchunk = NTOK / 256;
  for (int chunk = 0; chunk < nchunk; ++chunk) {
    const int i = chunk * 256 + t;
    const int myc = cat[i] & 63;
    ccat[t] = myc;
    __syncthreads();
    int rank = 0;
    for (int j = 0; j < t; ++j) rank += (ccat[j] == myc) ? 1 : 0;
    ord[running[myc] + rank] = i;
    __syncthreads();
    if (t < 64) {
      int cnt = 0;
      for (int j = 0; j < 256; ++j) cnt += (ccat[j] == t) ? 1 : 0;
      running[t] += cnt;
    }
    __syncthreads();
  }
}

// ---------------------------------------------------------------------------
// Kernel 2: per-(batch, group, head) 128x128 attention, d=32, fp32 WMMA.
// Gathers Q/K/V by sorted order, writes O scattered back to original order
// (fusing the inverse permutation). 128 threads = 4 wave32s.
// Dynamic LDS: Qs/Ks/Vs 128x36 fp32 + S 128x132 fp32 = 122880 B (<320KB/WGP).
// ---------------------------------------------------------------------------
__global__ __launch_bounds__(128) void acmsa_attn_kernel(
    const float* __restrict__ qkv, const int* __restrict__ order,
    const float* __restrict__ logit_scale, float* __restrict__ X) {
  extern __shared__ float smem[];
  float* Qs = smem;                  // [128][QSTR]
  float* Ks = Qs + GSIZE * QSTR;     // [128][QSTR]
  float* Vs = Ks + GSIZE * QSTR;     // [128][QSTR]
  float* S  = Vs + GSIZE * QSTR;     // [128][SSTR]

  const int gi = blockIdx.x;         // group (sorted-position / 128)
  const int head = blockIdx.y;
  const int bb = blockIdx.z;
  const int t = threadIdx.x;

  const int* ord = order + (size_t)bb * NTOK + gi * GSIZE;

  // --- gather Q,K,V tile rows (one token per thread), float4 vector loads ---
  {
    const int i = ord[t];
    const float* base = qkv + ((size_t)bb * NTOK + i) * (3 * CDIM) + head * DHEAD;
    const float4* q4 = reinterpret_cast<const float4*>(base);
    const float4* k4 = reinterpret_cast<const float4*>(base + CDIM);
    const float4* v4 = reinterpret_cast<const float4*>(base + 2 * CDIM);
    float4* qd = reinterpret_cast<float4*>(Qs + t * QSTR);
    float4* kd = reinterpret_cast<float4*>(Ks + t * QSTR);
    float4* vd = reinterpret_cast<float4*>(Vs + t * QSTR);
#pragma unroll
    for (int j = 0; j < DHEAD / 4; ++j) {
      qd[j] = q4[j];
      kd[j] = k4[j];
      vd[j] = v4[j];
    }
  }
  __syncthreads();

  const float scale =
      __expf(fminf(logit_scale[0], 4.6051701859880913680f /* ln(100) */));

  const int lane = t & 31;
  const int wv = t >> 5;        // wave id 0..3 -> row-tiles {2wv, 2wv+1}
  const int m = lane & 15;      // row/col within 16 (A:M, B:N, C:N)
  const int g = lane >> 4;      // lane-group (selects K/M halves)

  // --- S = (Q K^T) * scale ---
  // Preload the wave's A fragments once (2 row-tiles x 8 k-steps).
  v2f Afrag[2][8];
#pragma unroll
  for (int tmi = 0; tmi < 2; ++tmi) {
    const float* qrow = Qs + (16 * (2 * wv + tmi) + m) * QSTR;
#pragma unroll
    for (int kk = 0; kk < DHEAD / 4; ++kk) {
      const int kb = 4 * kk + 2 * g;
      Afrag[tmi][kk].x = qrow[kb];
      Afrag[tmi][kk].y = qrow[kb + 1];
    }
  }
  for (int tn = 0; tn < 8; ++tn) {
    const float* krow = Ks + (16 * tn + m) * QSTR;
    v8f acc0 = V8F_ZERO, acc1 = V8F_ZERO;
#pragma unroll
    for (int kk = 0; kk < DHEAD / 4; ++kk) {
      const int kb = 4 * kk + 2 * g;
      v2f bv;                       // B[k][n] = K[n][k]
      bv.x = krow[kb];
      bv.y = krow[kb + 1];
      acc0 = wmma4(Afrag[0][kk], bv, acc0);   // independent accumulators:
      acc1 = wmma4(Afrag[1][kk], bv, acc1);   // back-to-back WMMA, no RAW
    }
#pragma unroll
    for (int r = 0; r < 8; ++r) {
      S[(32 * wv + r + 8 * g) * SSTR + 16 * tn + m] = acc0[r] * scale;
      S[(32 * wv + 16 + r + 8 * g) * SSTR + 16 * tn + m] = acc1[r] * scale;
    }
  }
  __syncthreads();

  // --- softmax: one row per thread, float4 LDS ops ---
  {
    float4* row4 = reinterpret_cast<float4*>(S + t * SSTR);
    float mx = -3.0e38f;
#pragma unroll 4
    for (int j = 0; j < GSIZE / 4; ++j) {
      const float4 v = row4[j];
      mx = fmaxf(mx, fmaxf(fmaxf(v.x, v.y), fmaxf(v.z, v.w)));
    }
    float sum = 0.f;
#pragma unroll 4
    for (int j = 0; j < GSIZE / 4; ++j) {
      float4 v = row4[j];
      v.x = __expf(v.x - mx);
      v.y = __expf(v.y - mx);
      v.z = __expf(v.z - mx);
      v.w = __expf(v.w - mx);
      sum += (v.x + v.y) + (v.z + v.w);
      row4[j] = v;
    }
    const float inv = 1.f / sum;
#pragma unroll 4
    for (int j = 0; j < GSIZE / 4; ++j) {
      float4 v = row4[j];
      v.x *= inv; v.y *= inv; v.z *= inv; v.w *= inv;
      row4[j] = v;
    }
  }
  __syncthreads();

  // --- O = P V: 4 resident tiles per wave, one K sweep, 4 WMMAs/k-step ---
  {
    const float* p0 = S + (32 * wv + m) * SSTR;
    const float* p1 = S + (32 * wv + 16 + m) * SSTR;
    v8f o00 = V8F_ZERO, o01 = V8F_ZERO, o10 = V8F_ZERO, o11 = V8F_ZERO;
#pragma unroll 8
    for (int kk = 0; kk < GSIZE / 4; ++kk) {
      const int kb = 4 * kk + 2 * g;
      v2f a0, a1, b0, b1;
      a0.x = p0[kb];     a0.y = p0[kb + 1];
      a1.x = p1[kb];     a1.y = p1[kb + 1];
      b0.x = Vs[kb * QSTR + m];            b0.y = Vs[(kb + 1) * QSTR + m];
      b1.x = Vs[kb * QSTR + 16 + m];       b1.y = Vs[(kb + 1) * QSTR + 16 + m];
      o00 = wmma4(a0, b0, o00);
      o01 = wmma4(a0, b1, o01);
      o10 = wmma4(a1, b0, o10);
      o11 = wmma4(a1, b1, o11);
    }
    // scatter-store to original token order (one order[] read per row)
#pragma unroll
    for (int r = 0; r < 8; ++r) {
      const int rl0 = 32 * wv + r + 8 * g;
      const int rl1 = rl0 + 16;
      const size_t base0 =
          ((size_t)bb * NTOK + ord[rl0]) * CDIM + head * DHEAD + m;
      const size_t base1 =
          ((size_t)bb * NTOK + ord[rl1]) * CDIM + head * DHEAD + m;
      X[base0] = o00[r];
      X[base0 + 16] = o01[r];
      X[base1] = o10[r];
      X[base1 + 16] = o11[r];
    }
  }
}

// ---------------------------------------------------------------------------
// Kernel 3: out = X @ W^T + bias.  256 threads = 8 wave32s; 128 rows x 192
// cols per block; K tiled by 32 through LDS; 12 persistent accumulators/wave;
// one A fragment fanned out across all 12 column tiles per k-step.
// ---------------------------------------------------------------------------
__global__ __launch_bounds__(256) void acmsa_proj_kernel(
    const float* __restrict__ X, const float* __restrict__ W,
    const float* __restrict__ bias, float* __restrict__ out) {
  __shared__ float Xs[128 * QSTR];   // 128 x 32 (stride 36)
  __shared__ float Ws[192 * QSTR];   // 192 x 32 (stride 36)

  const int t = threadIdx.x;
  const int lane = t & 31;
  const int wv = t >> 5;             // wave 0..7 -> rows [16wv, 16wv+16)
  const int m = lane & 15;
  const int g = lane >> 4;
  const size_t rowBase = (size_t)blockIdx.x * 128;

  v8f acc[12];
#pragma unroll
  for (int j = 0; j < 12; ++j) acc[j] = V8F_ZERO;

  for (int k0 = 0; k0 < CDIM; k0 += 32) {
    __syncthreads();
    {
      const int row = t >> 1, half = (t & 1) * 16;     // 128 rows x two halves
      const float4* src =
          reinterpret_cast<const float4*>(X + (rowBase + row) * CDIM + k0 + half);
      float4* dst = reinterpret_cast<float4*>(Xs + row * QSTR + half);
#pragma unroll
      for (int j = 0; j < 4; ++j) dst[j] = src[j];
    }
    for (int jj = t; jj < 192 * 2; jj += 256) {        // 192 rows x two halves
      const int row = jj >> 1, half = (jj & 1) * 16;
      const float4* src =
          reinterpret_cast<const float4*>(W + (size_t)row * CDIM + k0 + half);
      float4* dst = reinterpret_cast<float4*>(Ws + row * QSTR + half);
#pragma unroll
      for (int j = 0; j < 4; ++j) dst[j] = src[j];
    }
    __syncthreads();

    const float* xrow = Xs + (16 * wv + m) * QSTR;
#pragma unroll
    for (int kk = 0; kk < 8; ++kk) {
      const int kb = 4 * kk + 2 * g;
      v2f a;                         // one A fragment per k-step...
      a.x = xrow[kb];
      a.y = xrow[kb + 1];
#pragma unroll
      for (int tn = 0; tn < 12; ++tn) {   // ...fanned across 12 tiles
        v2f bv;                      // B[k][o] = W[o][k]
        bv.x = Ws[(16 * tn + m) * QSTR + kb];
        bv.y = Ws[(16 * tn + m) * QSTR + kb + 1];
        acc[tn] = wmma4(a, bv, acc[tn]);
      }
    }
  }

#pragma unroll
  for (int tn = 0; tn < 12; ++tn) {
    const int o = 16 * tn + m;
    const float bo = bias[o];
#pragma unroll
    for (int r = 0; r < 8; ++r) {
      const int row = 16 * wv + r + 8 * g;
      out[(rowBase + row) * CDIM + o] = acc[tn][r] + bo;
    }
  }
}

// ---------------------------------------------------------------------------
extern "C" void kernel_launch(void* const* d_in, const int* in_sizes, int n_in,
                              void* d_out, int out_size, void* d_ws,
                              size_t ws_size, hipStream_t stream) {
  (void)in_sizes; (void)n_in; (void)out_size; (void)ws_size;
  const float* qkv = (const float*)d_in[0];
  const int* indices = (const int*)d_in[1];
  const float* proj_w = (const float*)d_in[2];
  const float* proj_b = (const float*)d_in[3];
  const float* logit_scale = (const float*)d_in[4];

  // workspace: X (b*n*c fp32) | order (b*n int32)
  float* X = (float*)d_ws;
  int* order =
      (int*)((char*)d_ws + (size_t)NBATCH * NTOK * CDIM * sizeof(float));

  acmsa_sort_kernel<<<NBATCH, 256, 0, stream>>>(indices, order);

  dim3 gridA(NTOK / GSIZE, NHEADS, NBATCH);
  const size_t shA = (size_t)(3 * GSIZE * QSTR + GSIZE * SSTR) * sizeof(float);
  acmsa_attn_kernel<<<gridA, GSIZE, shA, stream>>>(qkv, order, logit_scale, X);

  acmsa_proj_kernel<<<(NBATCH * NTOK) / 128, 256, 0, stream>>>(
      X, proj_w, proj_b, (float*)d_out);
}